// EdgeBuilder_78915729096712
// MI455X (gfx1250) — compile-verified
//
#include <hip/hip_runtime.h>
#include <hip/hip_bf16.h>
#include <cstdint>

typedef __attribute__((ext_vector_type(2))) float v2f;
typedef __attribute__((ext_vector_type(4))) float v4f;
typedef __attribute__((ext_vector_type(8))) float v8f;

#define N_NODES 1024
#define KSEL 16
#define ROWS_PER_BLK 16
#define EPSF 1e-12f

__device__ __forceinline__ uint32_t umin32(uint32_t a, uint32_t b) { return a < b ? a : b; }

// One workgroup = (batch b, 16-row block rb). 256 threads = 8 waves (wave32).
// Phase 0: per-node (|x|^2 or +INF if masked) -> LDS; mask folded into norms.
// Phase 1: 8 waves x 8 16x16 Gram tiles with V_WMMA_F32_16X16X4_F32;
//          d = sqrt(max(rs+cs-2*gram, eps)), diagonal -> +INF; strip to LDS.
// Phase 2: one wave per row: 16x extract-min on index-packed uint keys
//          (uniform poison via readfirstlane), then non-temporal dense output.
__global__ __launch_bounds__(256) void edge_knn_kernel(
    const float* __restrict__ nodes,
    const unsigned char* __restrict__ mask,
    float* __restrict__ out_edges,
    float* __restrict__ out_adj)
{
    extern __shared__ char smem[];
    float* lds_dist = (float*)smem;                                // 16*1024 f32
    float* lds_sq   = (float*)(smem + ROWS_PER_BLK * N_NODES * 4); // 1024 f32 (masked norms)

    const int rb   = blockIdx.x;   // row block 0..63
    const int b    = blockIdx.y;   // batch 0..63
    const int t    = threadIdx.x;
    const int wave = t >> 5;
    const int lane = t & 31;

    // ---- Phase 0: masked squared norms into LDS (invalid node -> +INF) ----
    for (int i = 0; i < 4; ++i) {
        int node = i * 256 + t;
        const v4f* p = (const v4f*)(nodes + ((size_t)(b * N_NODES + node) << 6)); // 64 f32/node
        float s = 0.f;
        #pragma unroll
        for (int k = 0; k < 8; ++k) {           // first 32 coords
            v4f x = p[k];
            s += x.x * x.x + x.y * x.y + x.z * x.z + x.w * x.w;
        }
        lds_sq[node] = mask[b * N_NODES + node] ? s : __builtin_inff();
    }
    __syncthreads();

    // ---- Phase 1: WMMA Gram -> distances ----
    const int hi = lane >> 4;
    const int ml = lane & 15;

    // A operand (16 rows x K=32), 8 chunks of K=4; lane L: (M=L%16, K=4c+2*hi..+1)
    v2f areg[8];
    {
        const int rowg = rb * ROWS_PER_BLK + ml;
        const float* pa = nodes + ((size_t)(b * N_NODES + rowg) << 6);
        #pragma unroll
        for (int c = 0; c < 8; ++c)
            areg[c] = *(const v2f*)(pa + 4 * c + 2 * hi);
    }

    // Row-side invariants for this lane's 8 output rows (masked norms).
    float rowsq[8];
    #pragma unroll
    for (int j = 0; j < 8; ++j)
        rowsq[j] = lds_sq[rb * ROWS_PER_BLK + j + 8 * hi];

    for (int tct = 0; tct < 8; ++tct) {
        const int ct   = wave + tct * 8;
        const int colg = ct * 16 + ml;
        const float* pb = nodes + ((size_t)(b * N_NODES + colg) << 6);

        v2f breg[8];
        #pragma unroll
        for (int c = 0; c < 8; ++c)
            breg[c] = *(const v2f*)(pb + 4 * c + 2 * hi);

        v8f acc = {};
        #pragma unroll
        for (int c = 0; c < 8; ++c)
            acc = __builtin_amdgcn_wmma_f32_16x16x4_f32(
                false, areg[c], false, breg[c], (short)0, acc, false, false);

        const float sqc = lds_sq[colg];       // +INF if column masked
        #pragma unroll
        for (int j = 0; j < 8; ++j) {
            const int m    = j + 8 * hi;      // C/D layout: reg j -> M=j / j+8
            const int rowg = rb * ROWS_PER_BLK + m;
            const float d2 = __builtin_fmaf(-2.0f, acc[j], rowsq[j] + sqc); // INF-propagates masks
            float d = __builtin_amdgcn_sqrtf(fmaxf(d2, EPSF));
            d = (rowg == colg) ? __builtin_inff() : d;                      // exclude self
            lds_dist[m * N_NODES + colg] = d;
        }
    }
    __syncthreads();

    // ---- Phase 2: per-row top-16 + streaming output ----
    for (int rr = 0; rr < 2; ++rr) {
        const int lr   = wave + rr * 8;
        const int rowg = rb * ROWS_PER_BLK + lr;
        const float* drow = lds_dist + lr * N_NODES;

        // Keys: positive-float bits (uint-comparable) with lane-local slot j
        // packed in the 5 LSBs; lane owns cols j*32+lane.
        uint32_t k[32];
        #pragma unroll
        for (int j = 0; j < 32; ++j)
            k[j] = (__float_as_uint(drow[j * 32 + lane]) & ~31u) | (uint32_t)j;

        for (int it = 0; it < KSEL; ++it) {
            uint32_t lmin = k[0];
            #pragma unroll
            for (int j = 1; j < 32; ++j) lmin = umin32(lmin, k[j]);

            uint32_t wm = lmin;
            #pragma unroll
            for (int off = 16; off > 0; off >>= 1)
                wm = umin32(wm, (uint32_t)__shfl_xor((int)wm, off, 32));
            wm = (uint32_t)__builtin_amdgcn_readfirstlane((int)wm);  // wave-uniform

            const int jm = (int)(wm & 31u);                          // uniform slot index
            #pragma unroll
            for (int j = 0; j < 32; ++j)
                if (j == jm)                                         // scalar-guarded
                    k[j] = (k[j] == wm) ? 0xFFFFFFFFu : k[j];        // only owner lane matches
        }

        const size_t obase = ((size_t)b * N_NODES + rowg) * (size_t)N_NODES;
        #pragma unroll 4
        for (int j = 0; j < 32; ++j) {
            const int   col = j * 32 + lane;
            const float d   = drow[col];
            const bool  s   = (k[j] == 0xFFFFFFFFu) && (d < 3.0e38f);
            __builtin_nontemporal_store(s ? d : 0.0f,    out_edges + obase + col);
            __builtin_nontemporal_store(s ? 1.0f : 0.0f, out_adj   + obase + col);
        }
    }
}

extern "C" void kernel_launch(void* const* d_in, const int* in_sizes, int n_in,
                              void* d_out, int out_size, void* d_ws, size_t ws_size,
                              hipStream_t stream) {
    (void)in_sizes; (void)n_in; (void)out_size; (void)d_ws; (void)ws_size;

    const float*         nodes = (const float*)d_in[0];          // [64,1024,64] f32
    const unsigned char* maskp = (const unsigned char*)d_in[1];  // [64,1024] bool

    float* out_edges = (float*)d_out;                            // [64,1024,1024] f32
    float* out_adj   = out_edges + (size_t)64 * 1024 * 1024;     // [64,1024,1024] bool-as-f32

    const dim3 grid(N_NODES / ROWS_PER_BLK, 64);
    const dim3 block(256);
    const size_t shmem = (size_t)ROWS_PER_BLK * N_NODES * 4      // dist strip: 64 KB
                       + (size_t)N_NODES * 4;                    // masked norms

    edge_knn_kernel<<<grid, block, shmem, stream>>>(nodes, maskp, out_edges, out_adj);
}